// BEVWarpAndAccumulate_79611513799004
// MI455X (gfx1250) — compile-verified
//
#include <hip/hip_runtime.h>
#include <math.h>

// ---------------------------------------------------------------------------
// Problem constants (match reference)
// ---------------------------------------------------------------------------
#define BB   4096
#define TT   4
#define DD   384
#define PP   8
#define CC   96          // D/4
#define N1   6144        // P*P*C
#define MROWS (BB*TT)    // 16384

typedef __attribute__((ext_vector_type(2))) float v2f;
typedef __attribute__((ext_vector_type(8))) float v8f;

// fp32 WMMA 16x16x4: D = A(16x4) * B(4x16) + C(16x16)
static __device__ __forceinline__ v8f wmma_f32(v2f a, v2f b, v8f c) {
  return __builtin_amdgcn_wmma_f32_16x16x4_f32(
      /*neg_a=*/false, a, /*neg_b=*/false, b,
      /*c_mod=*/(short)0, c, /*reuse_a=*/false, /*reuse_b=*/false);
}

static __device__ __forceinline__ float geluf(float x) {
  // exact erf GELU (reference uses approximate=False)
  return 0.5f * x * (1.0f + erff(x * 0.70710678118654752440f));
}

// ---------------------------------------------------------------------------
// Kernel 1: per-(b,t) 64-pixel warp+pool weight vectors.
// mean(grid_sample(spatial)) == sum_p w[b,t,p]*spatial[b,t,c,p]; w is channel-
// independent, so warp+pool collapses to one 64-long dot per (b,t).
// ---------------------------------------------------------------------------
__global__ void k_warp_weights(const float* __restrict__ ego,
                               float* __restrict__ wvec) {
  const int bt = blockIdx.x;            // 0..B*T-1, row = b*T + t
  const int b  = bt >> 2;
  const int t  = bt & 3;
  const int tid = threadIdx.x;          // 0..63 (one output pixel)
  __shared__ float acc[64];

  if (t == TT - 1) {                    // last frame: identity -> uniform mean
    wvec[(size_t)bt * 64 + tid] = 1.0f / 64.0f;
    return;
  }
  acc[tid] = 0.0f;
  __syncthreads();

  const int e = (b * (TT - 1) + t) * 3;
  const float dx = ego[e + 0], dy = ego[e + 1], dyaw = ego[e + 2];
  const float ca = cosf(dyaw), sa = sinf(dyaw);
  const float tx = dx * 0.02f, ty = dy * 0.02f;

  const int qx = tid & 7, qy = tid >> 3;
  const float xg = (qx + 0.5f) * 0.25f - 1.0f;   // align_corners=False grid
  const float yg = (qy + 0.5f) * 0.25f - 1.0f;
  const float gx =  ca * xg + sa * yg + tx;
  const float gy = -sa * xg + ca * yg + ty;
  const float x = (gx + 1.0f) * 4.0f - 0.5f;     // (gx+1)*(W/2)-0.5, W=8
  const float y = (gy + 1.0f) * 4.0f - 0.5f;
  const float x0f = floorf(x), y0f = floorf(y);
  const float wx1 = x - x0f, wy1 = y - y0f;
  const float wx0 = 1.0f - wx1, wy0 = 1.0f - wy1;
  const int x0 = (int)x0f, y0 = (int)y0f;

  #pragma unroll
  for (int dyi = 0; dyi < 2; ++dyi) {
    #pragma unroll
    for (int dxi = 0; dxi < 2; ++dxi) {
      const int xi = x0 + dxi, yi = y0 + dyi;
      const float w = (dyi ? wy1 : wy0) * (dxi ? wx1 : wx0);
      if (xi >= 0 && xi < 8 && yi >= 0 && yi < 8)
        atomicAdd(&acc[yi * 8 + xi], w);         // ds_add_f32
    }
  }
  __syncthreads();
  wvec[(size_t)bt * 64 + tid] = acc[tid] * (1.0f / 64.0f);
}

// ---------------------------------------------------------------------------
// Kernel 2: fused GEMM1 (16384x6144, K=384) + bias + GELU + weighted pool.
// Block: 256 thr (8 waves) handles 32 rows x all 96 channels. Two 16-row
// M-tiles share every B fragment -> W1 L2 traffic and vmem per WMMA halved.
// ---------------------------------------------------------------------------
#define LDA1 388   // 384 + 4 pad: multiple of 4 (float4 LDS fills stay aligned)

__global__ void __launch_bounds__(256)
k_gemm1_pool(const float* __restrict__ hv, const float* __restrict__ W1,
             const float* __restrict__ b1, const float* __restrict__ wvec,
             float* __restrict__ pooled_sp) {
  __shared__ float As[32 * LDA1];      // 48.5 KB
  __shared__ float Ws[32 * 64];        //  8   KB
  const int tid = threadIdx.x;
  const size_t rowBase = (size_t)blockIdx.x * 32;

  // float4-vectorized tile fills
  for (int i = tid; i < 32 * (DD / 4); i += 256) {
    const int r = i / (DD / 4), c4 = (i - r * (DD / 4)) * 4;
    *(float4*)&As[r * LDA1 + c4] = *(const float4*)&hv[(rowBase + r) * DD + c4];
  }
  for (int i = tid; i < 32 * 16; i += 256)
    *(float4*)&Ws[i * 4] = *(const float4*)&wvec[rowBase * 64 + i * 4];
  __syncthreads();

  const int lane = tid & 31;
  const int wave = tid >> 5;
  const int half = lane >> 4;          // 0: K=k..k+1, 1: K=k+2..k+3
  const int lcol = lane & 15;          // A row / B,D column within tile
  const int koff = half << 1;

  for (int ch = wave; ch < CC; ch += 8) {
    v8f c00 = {}, c01 = {}, c02 = {}, c03 = {};   // M-tile 0 (rows 0..15)
    v8f c10 = {}, c11 = {}, c12 = {}, c13 = {};   // M-tile 1 (rows 16..31)
    const float* wcol = W1 + ch * 64 + lcol;

    for (int k = 0; k < DD; k += 4) {
      const v2f a0 = *(const v2f*)&As[lcol * LDA1 + k + koff];
      const v2f a1 = *(const v2f*)&As[(16 + lcol) * LDA1 + k + koff];
      const float* bp = wcol + (size_t)(k + koff) * N1;
      v2f b;
      b.x = bp[0];  b.y = bp[N1];
      c00 = wmma_f32(a0, b, c00);  c10 = wmma_f32(a1, b, c10);
      b.x = bp[16]; b.y = bp[N1 + 16];
      c01 = wmma_f32(a0, b, c01);  c11 = wmma_f32(a1, b, c11);
      b.x = bp[32]; b.y = bp[N1 + 32];
      c02 = wmma_f32(a0, b, c02);  c12 = wmma_f32(a1, b, c12);
      b.x = bp[48]; b.y = bp[N1 + 48];
      c03 = wmma_f32(a0, b, c03);  c13 = wmma_f32(a1, b, c13);
    }

    const float bias0 = b1[ch * 64 +  0 + lcol];
    const float bias1 = b1[ch * 64 + 16 + lcol];
    const float bias2 = b1[ch * 64 + 32 + lcol];
    const float bias3 = b1[ch * 64 + 48 + lcol];

    float p0[8], p1[8];
    #pragma unroll
    for (int r = 0; r < 8; ++r) {
      const int m = r + 8 * half;      // row within each 16-row M-tile
      float s;
      s  = Ws[m * 64 +  0 + lcol] * geluf(c00[r] + bias0);
      s += Ws[m * 64 + 16 + lcol] * geluf(c01[r] + bias1);
      s += Ws[m * 64 + 32 + lcol] * geluf(c02[r] + bias2);
      s += Ws[m * 64 + 48 + lcol] * geluf(c03[r] + bias3);
      p0[r] = s;
      s  = Ws[(16 + m) * 64 +  0 + lcol] * geluf(c10[r] + bias0);
      s += Ws[(16 + m) * 64 + 16 + lcol] * geluf(c11[r] + bias1);
      s += Ws[(16 + m) * 64 + 32 + lcol] * geluf(c12[r] + bias2);
      s += Ws[(16 + m) * 64 + 48 + lcol] * geluf(c13[r] + bias3);
      p1[r] = s;
    }
    // reduce across the 16 lanes of each half (columns of the tile)
    #pragma unroll
    for (int mask = 1; mask <= 8; mask <<= 1) {
      #pragma unroll
      for (int r = 0; r < 8; ++r) {
        p0[r] += __shfl_xor(p0[r], mask, 32);
        p1[r] += __shfl_xor(p1[r], mask, 32);
      }
    }
    if (lcol == 0) {
      #pragma unroll
      for (int r = 0; r < 8; ++r) {
        pooled_sp[(rowBase + r + 8 * half) * CC + ch]      = p0[r];
        pooled_sp[(rowBase + 16 + r + 8 * half) * CC + ch] = p1[r];
      }
    }
  }
}

// ---------------------------------------------------------------------------
// Kernel 3: tok = gelu(pooled_sp @ W2 + b2); pooled[b] = sum_t softmax(tw)[t]*tok
// Block: 256 thr, 16 rows (= 4 batches x 4 frames), all 24 N-tiles. The 4
// frames of one batch share a lane in the fragment -> in-lane temporal sum.
// ---------------------------------------------------------------------------
#define LDA2 100   // 96 + 4 pad (multiple of 4)

__global__ void __launch_bounds__(256)
k_gemm2_temporal(const float* __restrict__ pooled_sp, const float* __restrict__ W2,
                 const float* __restrict__ b2, const float* __restrict__ tw,
                 float* __restrict__ pooled) {
  __shared__ float As[16 * LDA2];
  const int tid = threadIdx.x;
  const size_t rowBase = (size_t)blockIdx.x * 16;

  for (int i = tid; i < 16 * (CC / 4); i += 256) {
    const int r = i / (CC / 4), c4 = (i - r * (CC / 4)) * 4;
    *(float4*)&As[r * LDA2 + c4] = *(const float4*)&pooled_sp[(rowBase + r) * CC + c4];
  }
  __syncthreads();

  // softmax over 4 temporal weights (redundant per thread; trivial cost)
  float w0 = tw[0], w1 = tw[1], w2 = tw[2], w3 = tw[3];
  const float mx = fmaxf(fmaxf(w0, w1), fmaxf(w2, w3));
  w0 = __expf(w0 - mx); w1 = __expf(w1 - mx);
  w2 = __expf(w2 - mx); w3 = __expf(w3 - mx);
  const float inv = 1.0f / (w0 + w1 + w2 + w3);
  const float sw[4] = {w0 * inv, w1 * inv, w2 * inv, w3 * inv};

  const int lane = tid & 31;
  const int wave = tid >> 5;
  const int half = lane >> 4;
  const int lcol = lane & 15;
  const int koff = half << 1;

  for (int nt = wave; nt < DD / 16; nt += 8) {
    const int n0 = nt * 16;
    v8f acc = {};
    for (int k = 0; k < CC; k += 4) {
      const v2f a = *(const v2f*)&As[lcol * LDA2 + k + koff];
      const float* bp = W2 + (size_t)(k + koff) * DD + n0 + lcol;
      v2f b; b.x = bp[0]; b.y = bp[DD];
      acc = wmma_f32(a, b, acc);
    }
    const float bias = b2[n0 + lcol];
    float o0 = 0.0f, o1 = 0.0f;       // two batches per lane-half
    #pragma unroll
    for (int r = 0; r < 8; ++r) {
      const float v = geluf(acc[r] + bias) * sw[r & 3];
      if (r < 4) o0 += v; else o1 += v;
    }
    const size_t bg = (size_t)blockIdx.x * 4 + 2 * half;
    pooled[bg * DD + n0 + lcol]       = o0;
    pooled[(bg + 1) * DD + n0 + lcol] = o1;
  }
}

// ---------------------------------------------------------------------------
// Kernels 4/5: out = act(A @ W + bias), 4096x384 @ 384x384.
// ---------------------------------------------------------------------------
template <bool GELU>
__global__ void __launch_bounds__(256)
k_fuse(const float* __restrict__ A, const float* __restrict__ W,
       const float* __restrict__ bias, float* __restrict__ out) {
  __shared__ float As[16 * LDA1];
  const int tid = threadIdx.x;
  const size_t rowBase = (size_t)blockIdx.x * 16;

  for (int i = tid; i < 16 * (DD / 4); i += 256) {
    const int r = i / (DD / 4), c4 = (i - r * (DD / 4)) * 4;
    *(float4*)&As[r * LDA1 + c4] = *(const float4*)&A[(rowBase + r) * DD + c4];
  }
  __syncthreads();

  const int lane = tid & 31;
  const int wave = tid >> 5;
  const int half = lane >> 4;
  const int lcol = lane & 15;
  const int koff = half << 1;

  for (int nt = wave; nt < DD / 16; nt += 8) {
    const int n0 = nt * 16;
    v8f acc = {};
    for (int k = 0; k < DD; k += 4) {
      const v2f a = *(const v2f*)&As[lcol * LDA1 + k + koff];
      const float* bp = W + (size_t)(k + koff) * DD + n0 + lcol;
      v2f b; b.x = bp[0]; b.y = bp[DD];
      acc = wmma_f32(a, b, acc);
    }
    const float bv = bias[n0 + lcol];
    #pragma unroll
    for (int r = 0; r < 8; ++r) {
      float v = acc[r] + bv;
      if (GELU) v = geluf(v);
      out[(rowBase + r + 8 * half) * DD + n0 + lcol] = v;
    }
  }
}

// ---------------------------------------------------------------------------
// Launch
// ---------------------------------------------------------------------------
extern "C" void kernel_launch(void* const* d_in, const int* in_sizes, int n_in,
                              void* d_out, int out_size, void* d_ws, size_t ws_size,
                              hipStream_t stream) {
  (void)in_sizes; (void)n_in; (void)out_size; (void)ws_size;

  const float* hv  = (const float*)d_in[0];
  const float* ego = (const float*)d_in[1];
  const float* W1  = (const float*)d_in[2];
  const float* b1  = (const float*)d_in[3];
  const float* W2  = (const float*)d_in[4];
  const float* b2  = (const float*)d_in[5];
  const float* tw  = (const float*)d_in[6];
  const float* Wf1 = (const float*)d_in[7];
  const float* bf1 = (const float*)d_in[8];
  const float* Wf2 = (const float*)d_in[9];
  const float* bf2 = (const float*)d_in[10];
  float* out = (float*)d_out;

  float* wvec   = (float*)d_ws;                       // 16384*64  = 4 MB
  float* psp    = wvec + (size_t)MROWS * 64;          // 16384*96  = 6 MB
  float* pooled = psp  + (size_t)MROWS * CC;          // 4096*384  = 6 MB
  float* hmid   = pooled + (size_t)BB * DD;           // 4096*384  = 6 MB

  k_warp_weights   <<<MROWS,      64, 0, stream>>>(ego, wvec);
  k_gemm1_pool     <<<MROWS / 32, 256, 0, stream>>>(hv, W1, b1, wvec, psp);
  k_gemm2_temporal <<<MROWS / 16, 256, 0, stream>>>(psp, W2, b2, tw, pooled);
  k_fuse<true>     <<<BB / 16,    256, 0, stream>>>(pooled, Wf1, bf1, hmid);
  k_fuse<false>    <<<BB / 16,    256, 0, stream>>>(hmid, Wf2, bf2, out);
}